// GraphSAGEModel_89902255439931
// MI455X (gfx1250) — compile-verified
//
#include <hip/hip_runtime.h>

// GraphSAGE 2-layer on MI455X (gfx1250, wave32).
//
// Strategy:
//  * Edge phase (dominant): 1 wave per edge; 32 lanes x float2 = coalesced
//    256B row gather; scatter via global_atomic_add_f32 into an L2-resident
//    25.6MB accumulator (192MB L2 absorbs all RMW traffic). Degree computed
//    once (dst-only) and reused by layer 2.
//  * Node phase: fp32 WMMA (v_wmma_f32_16x16x4_f32), 1 wave per 16-node tile.
//    Weights staged transposed in LDS (stride 66: bank-conflict-free, 8B
//    aligned ds_load_b64 B-fragments). Accumulator carries agg@W_l + x@W_r,
//    then bias (+ReLU for layer 1) on store.
//
// Workspace layout (floats): deg[N] | agg1[N*64] | h[N*64] | agg2[N*64]
//   = 77.2 MB total.

typedef __attribute__((ext_vector_type(2))) float v2f;
typedef __attribute__((ext_vector_type(8))) float v8f;

#define N_NODES 100000
#define LDSTRIDE 66   // 64 + 2: even (8B-aligned b64 frags), step-2 banks

// ---------------------------------------------------------------------------
__global__ void zero_ws_kernel(float* __restrict__ p, size_t n) {
    size_t i = (size_t)blockIdx.x * blockDim.x + threadIdx.x;
    size_t stride = (size_t)gridDim.x * blockDim.x;
    for (; i < n; i += stride) p[i] = 0.0f;
}

// ---------------------------------------------------------------------------
// One wave per edge. Lane L moves columns {2L, 2L+1} of the 64-wide feature
// row: coalesced float2 gather of feat[src], atomic scatter into agg[dst].
template <bool COMPUTE_DEG>
__global__ void __launch_bounds__(256)
edge_agg_kernel(const float* __restrict__ feat,
                const int*   __restrict__ src,
                const int*   __restrict__ dst,
                float*       __restrict__ agg,
                float*       __restrict__ deg,
                int E) {
    int gid  = blockIdx.x * blockDim.x + threadIdx.x;
    int e    = gid >> 5;          // wave-uniform
    if (e >= E) return;
    int lane = gid & 31;

    int s = src[e];
    int d = dst[e];

    v2f v = *(const v2f*)(feat + (size_t)s * 64 + lane * 2);
    float* o = agg + (size_t)d * 64 + lane * 2;
    atomicAdd(o,     v.x);        // global_atomic_add_f32 (no return, L2)
    atomicAdd(o + 1, v.y);
    if (COMPUTE_DEG && lane == 0) atomicAdd(deg + d, 1.0f);
}

// ---------------------------------------------------------------------------
// Layer 1: h = relu( (agg1/deg) @ W1_l + b1 + x @ W1_r ), d_in=d_h=64.
// One wave per 16-node tile; 4 accumulators (16x16 each) cover the 64 cols.
__global__ void __launch_bounds__(256)
sage_layer1_wmma(const float* __restrict__ x,
                 const float* __restrict__ agg,
                 const float* __restrict__ deg,
                 const float* __restrict__ Wl,   // [64,64] row-major (k,n)
                 const float* __restrict__ bl,   // [64]
                 const float* __restrict__ Wr,   // [64,64]
                 float*       __restrict__ h,
                 int nTiles) {
    __shared__ float sWl[64 * LDSTRIDE];   // transposed: sW[n*66 + k]
    __shared__ float sWr[64 * LDSTRIDE];
    __shared__ float sb[64];

    for (int i = threadIdx.x; i < 64 * 64; i += 256) {
        int k = i >> 6, n = i & 63;
        sWl[n * LDSTRIDE + k] = Wl[i];
        sWr[n * LDSTRIDE + k] = Wr[i];
    }
    if (threadIdx.x < 64) sb[threadIdx.x] = bl[threadIdx.x];
    __syncthreads();

    int tile = blockIdx.x * 8 + (threadIdx.x >> 5);   // wave-uniform
    if (tile >= nTiles) return;                       // EXEC stays all-1s

    int lane = threadIdx.x & 31;
    int half = lane >> 4;      // 0: K={0,1} slice   1: K={2,3} slice
    int l    = lane & 15;      // A: row index; B/C: column index
    int m0   = tile * 16;
    int row  = m0 + l;

    float rdeg = 1.0f / fmaxf(deg[row], 1.0f);

    const float* xrow = x   + (size_t)row * 64 + half * 2;
    const float* arow = agg + (size_t)row * 64 + half * 2;

    v8f acc[4] = {};
#pragma unroll
    for (int ks = 0; ks < 16; ++ks) {
        int k = ks * 4;
        // A fragments (16x4 f32 layout): lane holds feat[row][k+2*half .. +1]
        v2f ax = *(const v2f*)(xrow + k);
        v2f ag = *(const v2f*)(arow + k);
        ag.x *= rdeg; ag.y *= rdeg;

        int bo = k + half * 2;
#pragma unroll
        for (int nt = 0; nt < 4; ++nt) {
            // B fragments (4x16): lane holds W[k+2*half .. +1][nt*16+l]
            v2f bL = *(const v2f*)(&sWl[(nt * 16 + l) * LDSTRIDE + bo]);
            acc[nt] = __builtin_amdgcn_wmma_f32_16x16x4_f32(
                false, ag, false, bL, (short)0, acc[nt], false, false);
            v2f bR = *(const v2f*)(&sWr[(nt * 16 + l) * LDSTRIDE + bo]);
            acc[nt] = __builtin_amdgcn_wmma_f32_16x16x4_f32(
                false, ax, false, bR, (short)0, acc[nt], false, false);
        }
    }

    // C/D layout: VGPR r -> row m0 + r + 8*half, col = nt*16 + l
#pragma unroll
    for (int r = 0; r < 8; ++r) {
        int rr = m0 + r + half * 8;
#pragma unroll
        for (int nt = 0; nt < 4; ++nt) {
            int col = nt * 16 + l;
            float v = acc[nt][r] + sb[col];
            h[(size_t)rr * 64 + col] = fmaxf(v, 0.0f);
        }
    }
}

// ---------------------------------------------------------------------------
// Layer 2: out = (agg2/deg) @ W2_l + b2 + h @ W2_r, d_out=16 (one n-tile).
__global__ void __launch_bounds__(256)
sage_layer2_wmma(const float* __restrict__ hin,
                 const float* __restrict__ agg,
                 const float* __restrict__ deg,
                 const float* __restrict__ Wl,   // [64,16]
                 const float* __restrict__ bl,   // [16]
                 const float* __restrict__ Wr,   // [64,16]
                 float*       __restrict__ out,
                 int nTiles) {
    __shared__ float sWl[16 * LDSTRIDE];
    __shared__ float sWr[16 * LDSTRIDE];
    __shared__ float sb[16];

    for (int i = threadIdx.x; i < 64 * 16; i += 256) {
        int k = i >> 4, n = i & 15;
        sWl[n * LDSTRIDE + k] = Wl[i];
        sWr[n * LDSTRIDE + k] = Wr[i];
    }
    if (threadIdx.x < 16) sb[threadIdx.x] = bl[threadIdx.x];
    __syncthreads();

    int tile = blockIdx.x * 8 + (threadIdx.x >> 5);
    if (tile >= nTiles) return;

    int lane = threadIdx.x & 31;
    int half = lane >> 4;
    int l    = lane & 15;
    int m0   = tile * 16;
    int row  = m0 + l;

    float rdeg = 1.0f / fmaxf(deg[row], 1.0f);

    const float* hrow = hin + (size_t)row * 64 + half * 2;
    const float* arow = agg + (size_t)row * 64 + half * 2;

    v8f acc = {};
#pragma unroll
    for (int ks = 0; ks < 16; ++ks) {
        int k = ks * 4;
        v2f ah = *(const v2f*)(hrow + k);
        v2f ag = *(const v2f*)(arow + k);
        ag.x *= rdeg; ag.y *= rdeg;

        int bo = k + half * 2;
        v2f bL = *(const v2f*)(&sWl[l * LDSTRIDE + bo]);
        acc = __builtin_amdgcn_wmma_f32_16x16x4_f32(
            false, ag, false, bL, (short)0, acc, false, false);
        v2f bR = *(const v2f*)(&sWr[l * LDSTRIDE + bo]);
        acc = __builtin_amdgcn_wmma_f32_16x16x4_f32(
            false, ah, false, bR, (short)0, acc, false, false);
    }

#pragma unroll
    for (int r = 0; r < 8; ++r) {
        int rr = m0 + r + half * 8;
        out[(size_t)rr * 16 + l] = acc[r] + sb[l];
    }
}

// ---------------------------------------------------------------------------
extern "C" void kernel_launch(void* const* d_in, const int* in_sizes, int n_in,
                              void* d_out, int out_size, void* d_ws, size_t ws_size,
                              hipStream_t stream) {
    const float* x   = (const float*)d_in[0];
    const int*   ei  = (const int*)d_in[1];   // edge_index [2, E] (int32)
    const float* W1l = (const float*)d_in[2];
    const float* b1  = (const float*)d_in[3];
    const float* W1r = (const float*)d_in[4];
    const float* W2l = (const float*)d_in[5];
    const float* b2  = (const float*)d_in[6];
    const float* W2r = (const float*)d_in[7];

    int E = in_sizes[1] / 2;
    const int* src = ei;
    const int* dst = ei + E;

    float* ws   = (float*)d_ws;
    float* deg  = ws;                                  // [N]
    float* agg1 = ws + N_NODES;                        // [N,64]
    float* hbuf = agg1 + (size_t)N_NODES * 64;         // [N,64]
    float* agg2 = hbuf + (size_t)N_NODES * 64;         // [N,64]
    float* out  = (float*)d_out;                       // [N,16]

    // Zero accumulators every call (atomic accumulation; deterministic).
    zero_ws_kernel<<<2048, 256, 0, stream>>>(deg, (size_t)N_NODES * 65);   // deg+agg1
    zero_ws_kernel<<<2048, 256, 0, stream>>>(agg2, (size_t)N_NODES * 64);

    int edgeBlocks = (int)(((long long)E * 32 + 255) / 256);
    int nTiles     = (N_NODES + 15) / 16;              // 6250 (exact)
    int gemmBlocks = (nTiles + 7) / 8;

    // Layer 1
    edge_agg_kernel<true><<<edgeBlocks, 256, 0, stream>>>(x, src, dst, agg1, deg, E);
    sage_layer1_wmma<<<gemmBlocks, 256, 0, stream>>>(x, agg1, deg, W1l, b1, W1r,
                                                     hbuf, nTiles);
    // Layer 2 (deg reused: depends only on dst)
    edge_agg_kernel<false><<<edgeBlocks, 256, 0, stream>>>(hbuf, src, dst, agg2, deg, E);
    sage_layer2_wmma<<<gemmBlocks, 256, 0, stream>>>(hbuf, agg2, deg, W2l, b2, W2r,
                                                     out, nTiles);
}